// GateNetwork_3298534884238
// MI455X (gfx1250) — compile-verified
//
#include <hip/hip_runtime.h>
#include <hip/hip_bf16.h>
#include <math.h>

// Sizes from the reference
#define BDIM 64
#define CDIM 512
#define HWSZ 4096
#define EDIM 16
#define KSEL 4

typedef __attribute__((ext_vector_type(2))) float v2f;
typedef __attribute__((ext_vector_type(8))) float v8f;

// ---------------------------------------------------------------------------
// Kernel 1: fused adaptive max + avg pool to 1x1.
// One 256-thread block per (b,c) row of 4096 contiguous floats.
// Bandwidth-bound: 512 MB read -> ~22 us at 23.3 TB/s is the whole budget.
// ---------------------------------------------------------------------------
__global__ __launch_bounds__(256) void pool_kernel(const float* __restrict__ x,
                                                   float* __restrict__ pooled) {
  const int row = blockIdx.x;                    // b*C + c
  const float4* p = reinterpret_cast<const float4*>(x + (size_t)row * HWSZ);
  const int t = threadIdx.x;                     // 0..255

  float mx = -__builtin_inff();
  float sm = 0.0f;
#pragma unroll
  for (int j = 0; j < 4; ++j) {
    float4 v = p[t + j * 256];                   // coalesced global_load_b128
    mx = fmaxf(mx, fmaxf(fmaxf(v.x, v.y), fmaxf(v.z, v.w)));
    sm += (v.x + v.y) + (v.z + v.w);
  }

  // wave32 reduction
#pragma unroll
  for (int m = 16; m >= 1; m >>= 1) {
    mx = fmaxf(mx, __shfl_xor(mx, m, 32));
    sm += __shfl_xor(sm, m, 32);
  }

  __shared__ float smx[8];
  __shared__ float ssm[8];
  const int wave = t >> 5;
  const int lane = t & 31;
  if (lane == 0) { smx[wave] = mx; ssm[wave] = sm; }
  __syncthreads();
  if (t == 0) {
    float M = smx[0], S = ssm[0];
#pragma unroll
    for (int i = 1; i < 8; ++i) { M = fmaxf(M, smx[i]); S += ssm[i]; }
    pooled[row] = M + S * (1.0f / (float)HWSZ);
  }
}

// ---------------------------------------------------------------------------
// Kernel 2: dual GEMM [64,512]x[512,16] via V_WMMA_F32_16X16X4_F32 (fp32 exact
// path), then per-row noise normalization, top-4 selection, masked softmax.
// 1 block x 4 waves; wave w owns batch rows 16w..16w+15.
//
// Fragment layouts (ISA 7.12.2, 32-bit, wave32):
//   A 16x4 : lane(l=lane&15, half=lane>>4): VGPR0 = A[l][k+2*half],
//            VGPR1 = A[l][k+2*half+1]            -> float2 load
//   B 4x16 : VGPR0 = B[k+2*half][l], VGPR1 = B[k+2*half+1][l]
//            with B[kk][n] = W[n][kk]            -> same float2 addressing
//   C 16x16: lane half 0: VGPR j = row j; half 1: VGPR j = row 8+j; col = l
// ---------------------------------------------------------------------------
__global__ __launch_bounds__(128) void gate_kernel(const float* __restrict__ pooled,
                                                   const float* __restrict__ B0w,
                                                   const float* __restrict__ b0,
                                                   const float* __restrict__ B1w,
                                                   const float* __restrict__ b1,
                                                   float* __restrict__ out) {
  const int lane = threadIdx.x & 31;
  const int wave = threadIdx.x >> 5;
  const int half = lane >> 4;
  const int l    = lane & 15;
  const int off  = 2 * half;

  const float* aBase  = pooled + (size_t)(16 * wave + l) * CDIM + off;
  const float* w0Base = B0w    + (size_t)l * CDIM + off;
  const float* w1Base = B1w    + (size_t)l * CDIM + off;

  v8f ch = {0.f, 0.f, 0.f, 0.f, 0.f, 0.f, 0.f, 0.f};  // fc0 accumulator
  v8f cn = {0.f, 0.f, 0.f, 0.f, 0.f, 0.f, 0.f, 0.f};  // fc1 accumulator

#pragma unroll 4
  for (int k = 0; k < CDIM; k += 4) {
    v2f a  = *(const v2f*)(aBase + k);
    v2f w0 = *(const v2f*)(w0Base + k);
    v2f w1 = *(const v2f*)(w1Base + k);
    ch = __builtin_amdgcn_wmma_f32_16x16x4_f32(false, a, false, w0, (short)0, ch,
                                               false, false);
    cn = __builtin_amdgcn_wmma_f32_16x16x4_f32(false, a, false, w1, (short)0, cn,
                                               false, false);
  }

  const float b0v = b0[l];
  const float b1v = b1[l];

#pragma unroll
  for (int j = 0; j < 8; ++j) {
    // fc0 + LeakyReLU(0.2)
    float hv = ch[j] + b0v;
    hv = (hv >= 0.0f) ? hv : 0.2f * hv;

    // fc1 + softplus, branchless stable form:
    //   softplus(t) = max(t,0) + log(1 + exp(-|t|)), log arg in (1,2]
    float tv = cn[j] + b1v;
    float nv = fmaxf(tv, 0.0f) + __logf(1.0f + __expf(-fabsf(tv)));

    // per-row mean / std (ddof=1) across the 16 experts (one 16-lane half)
    float sum = nv, sq = nv * nv;
#pragma unroll
    for (int m = 8; m >= 1; m >>= 1) {
      sum += __shfl_xor(sum, m, 32);
      sq  += __shfl_xor(sq,  m, 32);
    }
    const float mu  = sum * (1.0f / 16.0f);
    const float var = (sq - 16.0f * mu * mu) * (1.0f / 15.0f);
    const float sd  = sqrtf(fmaxf(var, 1e-20f));

    const float score = hv + (nv - mu) / sd;

    // rank within row: #lanes strictly greater, ties broken by lane index
    int r = 0;
    const int base = lane & 16;
#pragma unroll
    for (int t2 = 0; t2 < 16; ++t2) {
      float o = __shfl(score, base + t2, 32);
      r += (o > score) || ((o == score) && (t2 < l));
    }

    // mask non-top-K with -inf, softmax over the row
    float logit = (r < KSEL) ? hv : -__builtin_inff();
    float mxl = logit;
#pragma unroll
    for (int m = 8; m >= 1; m >>= 1) mxl = fmaxf(mxl, __shfl_xor(mxl, m, 32));
    float e = __expf(logit - mxl);                 // exp(-inf) == 0
    float s = e;
#pragma unroll
    for (int m = 8; m >= 1; m >>= 1) s += __shfl_xor(s, m, 32);

    const int rowM = 16 * wave + j + 8 * half;
    out[(size_t)rowM * EDIM + l] = e / s;
  }
}

// ---------------------------------------------------------------------------
extern "C" void kernel_launch(void* const* d_in, const int* in_sizes, int n_in,
                              void* d_out, int out_size, void* d_ws, size_t ws_size,
                              hipStream_t stream) {
  const float* x  = (const float*)d_in[0];   // [B,C,H,W]
  const float* W0 = (const float*)d_in[1];   // [E,C]
  const float* b0 = (const float*)d_in[2];   // [E]
  const float* W1 = (const float*)d_in[3];   // [E,C]
  const float* b1 = (const float*)d_in[4];   // [E]
  float* out    = (float*)d_out;             // [B,E]
  float* pooled = (float*)d_ws;              // [B,C] = 128 KB scratch

  pool_kernel<<<BDIM * CDIM, 256, 0, stream>>>(x, pooled);
  gate_kernel<<<1, 128, 0, stream>>>(pooled, W0, b0, W1, b1, out);
}